// GraphEncoder_88106959110337
// MI455X (gfx1250) — compile-verified
//
#include <hip/hip_runtime.h>
#include <math.h>

#define VNUM 50000
#define OUT  128
#define EPS  1e-7f

typedef __attribute__((ext_vector_type(2))) float v2f;
typedef __attribute__((ext_vector_type(8))) float v8f;

// ---------------- Kernel 1: zero the segment-sum accumulator ----------------
__global__ void k_zero(float* __restrict__ p, int n4) {
    int i      = blockIdx.x * blockDim.x + threadIdx.x;
    int stride = gridDim.x * blockDim.x;
    float4 z = make_float4(0.f, 0.f, 0.f, 0.f);
    for (; i < n4; i += stride) ((float4*)p)[i] = z;
}

// ---------------- Kernel 2: edge gather + scaled scatter-add ----------------
// One wave32 per edge (grid-stride). Lane l owns 4 contiguous channels:
// float4 gather (global_load_b128) + 4x global_atomic_add_f32 into the
// 25.6MB accumulator (L2-resident on MI455X).
__global__ void k_edges(const int* __restrict__ sidx, const int* __restrict__ tidx,
                        const float* __restrict__ enorm, const float* __restrict__ esgn,
                        const float* __restrict__ vrepr, float* __restrict__ acc,
                        int nedges) {
    const int lane   = threadIdx.x & 31;
    const int wave   = (blockIdx.x * blockDim.x + threadIdx.x) >> 5;
    const int nwaves = (gridDim.x * blockDim.x) >> 5;
    for (int e = wave; e < nedges; e += nwaves) {
        const float w = esgn[e] * enorm[e];
        const int   s = sidx[e];
        const int   t = tidx[e];
        const float4 v = *(const float4*)(vrepr + (size_t)s * OUT + lane * 4);
        float* dst = acc + (size_t)t * OUT + lane * 4;
        atomicAdd(dst + 0, v.x * w);
        atomicAdd(dst + 1, v.y * w);
        atomicAdd(dst + 2, v.z * w);
        atomicAdd(dst + 3, v.w * w);
    }
}

// ---------------- Kernel 3: dual GEMM + bias / softplus epilogue ------------
// One wave per 16-row tile. A = ptr tile (16x128 f32), B[k,n] = w[n,k].
// V_WMMA_F32_16X16X4_F32, 32 K-steps. Writes loc in-place over the ptr
// accumulator (A fully register-resident first) and std to the second half.
__global__ __launch_bounds__(256)
void k_gemm(const float* __restrict__ ptr_in,
            const float* __restrict__ loc_w, const float* __restrict__ loc_b,
            const float* __restrict__ std_w, const float* __restrict__ std_b,
            float* __restrict__ out) {
    const int lane = threadIdx.x & 31;
    const int half = lane >> 4;   // 0: K={0,1} / M rows 0-7 ; 1: K={2,3} / M rows 8-15
    const int lrow = lane & 15;
    const int tile = (blockIdx.x * blockDim.x + threadIdx.x) >> 5;
    if (tile >= VNUM / 16) return;          // wave-uniform: EXEC stays all-1s
    const int m0 = tile * 16;

    // Preload the full 16x128 A tile into registers (ISA A-layout for f32 16x4:
    // lanes 0-15 hold M=lane, VGPR0=K0/VGPR1=K1; lanes 16-31 hold K2/K3).
    const float* arow = ptr_in + (size_t)(m0 + lrow) * OUT + half * 2;
    v2f a[32];
#pragma unroll
    for (int kt = 0; kt < 32; ++kt)
        a[kt] = *(const v2f*)(arow + kt * 4);

    float* loc_out = out;
    float* std_out = out + (size_t)VNUM * OUT;

    for (int nt = 0; nt < 8; ++nt) {
        const int n = nt * 16 + lrow;       // global output channel for this lane

        // ---- loc = ptr @ loc_w.T + loc_b ----
        {
            const float* brow = loc_w + (size_t)n * OUT + half * 2;
            v8f acc = {};
#pragma unroll
            for (int kt = 0; kt < 32; ++kt) {
                v2f b = *(const v2f*)(brow + kt * 4);
                acc = __builtin_amdgcn_wmma_f32_16x16x4_f32(
                    /*neg_a=*/false, a[kt], /*neg_b=*/false, b,
                    /*c_mod=*/(short)0, acc, /*reuse_a=*/false, /*reuse_b=*/false);
            }
            const float bias = loc_b[n];
#pragma unroll
            for (int i = 0; i < 8; ++i) {
                const int row = m0 + half * 8 + i;
                loc_out[(size_t)row * OUT + n] = acc[i] + bias;
            }
        }

        // ---- std = softplus(ptr @ std_w.T + std_b) + EPS ----
        {
            const float* brow = std_w + (size_t)n * OUT + half * 2;
            v8f acc = {};
#pragma unroll
            for (int kt = 0; kt < 32; ++kt) {
                v2f b = *(const v2f*)(brow + kt * 4);
                acc = __builtin_amdgcn_wmma_f32_16x16x4_f32(
                    false, a[kt], false, b, (short)0, acc, false, false);
            }
            const float bias = std_b[n];
#pragma unroll
            for (int i = 0; i < 8; ++i) {
                const int row = m0 + half * 8 + i;
                const float x  = acc[i] + bias;
                const float sp = (x > 20.0f) ? x : log1pf(expf(x));
                std_out[(size_t)row * OUT + n] = sp + EPS;
            }
        }
    }
}

extern "C" void kernel_launch(void* const* d_in, const int* in_sizes, int n_in,
                              void* d_out, int out_size, void* d_ws, size_t ws_size,
                              hipStream_t stream) {
    const int*   sidx  = (const int*)  d_in[0];
    const int*   tidx  = (const int*)  d_in[1];
    const float* enorm = (const float*)d_in[2];
    const float* esgn  = (const float*)d_in[3];
    const float* vrepr = (const float*)d_in[4];
    const float* loc_w = (const float*)d_in[5];
    const float* loc_b = (const float*)d_in[6];
    const float* std_w = (const float*)d_in[7];
    const float* std_b = (const float*)d_in[8];

    float* out = (float*)d_out;
    float* acc = out;                       // loc half doubles as ptr accumulator
    const int nedges = in_sizes[0];

    // 1) zero accumulator (VNUM*OUT floats = 1.6M float4s)
    const int n4 = VNUM * OUT / 4;
    k_zero<<<2048, 256, 0, stream>>>(acc, n4);

    // 2) edge gather/scatter: 4096 blocks * 8 waves, ~49 edges per wave
    k_edges<<<4096, 256, 0, stream>>>(sidx, tidx, enorm, esgn, vrepr, acc, nedges);

    // 3) dual WMMA GEMM + epilogue: 3125 row-tiles, 8 waves/block
    const int ntiles = VNUM / 16;
    k_gemm<<<(ntiles + 7) / 8, 256, 0, stream>>>(acc, loc_w, loc_b, std_w, std_b, out);
}